// GINLayer_11587821765006
// MI455X (gfx1250) — compile-verified
//
#include <hip/hip_runtime.h>

#define N_NODES 100000
#define N_EDGES 1600000
#define D 128
#define LDS_STRIDE 132   // 132 dwords/row -> lanes hit distinct LDS banks on b64 reads

typedef __attribute__((ext_vector_type(2))) float v2f;
typedef __attribute__((ext_vector_type(8))) float v8f;

// ---------------------------------------------------------------------------
// Kernel 1: out = (1 + eps) * x      (folds the GIN self-term into the
// accumulation buffer so the scatter kernel can atomically add on top)
// ---------------------------------------------------------------------------
__global__ __launch_bounds__(256) void gin_init(const float* __restrict__ x,
                                                const float* __restrict__ eps,
                                                float* __restrict__ out) {
    const float s = 1.0f + eps[0];
    size_t i = (size_t)blockIdx.x * blockDim.x + threadIdx.x;  // float4 index
    const float4* __restrict__ x4 = (const float4*)x;
    float4* __restrict__ o4 = (float4*)out;
    float4 v = x4[i];
    v.x *= s; v.y *= s; v.z *= s; v.w *= s;
    o4[i] = v;
}

// ---------------------------------------------------------------------------
// Kernel 2: scatter-add edge messages. One wave32 per edge: each lane loads
// 4 floats of x[src] (global_load_b128) and does 4 f32 atomics into out[dst].
// x and out both live in the 192MB L2, so these atomics never touch HBM.
// ---------------------------------------------------------------------------
__global__ __launch_bounds__(256) void gin_scatter(const float* __restrict__ x,
                                                   const long long* __restrict__ ei,
                                                   float* __restrict__ out) {
    const int lane = threadIdx.x & 31;
    const size_t e = (size_t)blockIdx.x * 8 + (threadIdx.x >> 5);  // 8 waves/block
    const long long src = ei[e];
    const long long dst = ei[(size_t)N_EDGES + e];

    const float4* __restrict__ xs = (const float4*)(x + (size_t)src * D);
    float4 v = xs[lane];

    float* o = out + (size_t)dst * D + lane * 4;
    unsafeAtomicAdd(o + 0, v.x);
    unsafeAtomicAdd(o + 1, v.y);
    unsafeAtomicAdd(o + 2, v.z);
    unsafeAtomicAdd(o + 3, v.w);
}

// ---------------------------------------------------------------------------
// Kernel 3: fused 2-layer MLP, in-place on out.
// Block = 256 threads = 8 wave32s; block owns 16 rows, wave w owns N-columns
// [16w, 16w+16). Per layer: 32 x V_WMMA_F32_16X16X4_F32 (full f32 precision).
//
// Fragment layouts (ISA 7.12.2, f32):
//   A 16x4 : lanes 0-15 row M=lane, {v0,v1} = K{0,1}; lanes 16-31 = K{2,3}
//   B 4x16 : v0 lanes0-15 = row K0, lanes16-31 = row K2; v1 = K1 / K3
//   C 16x16: vgpr r, lanes0-15 -> (M=r, N=lane); lanes16-31 -> (M=r+8)
// ---------------------------------------------------------------------------
__global__ __launch_bounds__(256) void gin_mlp(float* __restrict__ out,
                                               const float* __restrict__ W1,
                                               const float* __restrict__ b1,
                                               const float* __restrict__ W2,
                                               const float* __restrict__ b2) {
    __shared__ float s_in[16 * LDS_STRIDE];
    __shared__ float s_h[16 * LDS_STRIDE];

    const int tid  = threadIdx.x;
    const int wave = tid >> 5;        // 0..7 -> N tile
    const int lane = tid & 31;
    const int half = lane >> 4;       // 0 or 1 (selects K/M half per ISA layout)
    const int l16  = lane & 15;
    const size_t rowbase = (size_t)blockIdx.x * 16;

    // Cooperative load of the 16x128 input tile into padded LDS (coalesced).
    {
        const int r = tid >> 4;            // 0..15
        const int c = (tid & 15) * 8;      // 0,8,...,120
        const float4* __restrict__ g = (const float4*)(out + (rowbase + r) * D + c);
        float4 v0 = g[0];
        float4 v1 = g[1];
        *(float4*)&s_in[r * LDS_STRIDE + c]     = v0;
        *(float4*)&s_in[r * LDS_STRIDE + c + 4] = v1;
    }
    __syncthreads();

    const int cn = wave * 16 + l16;   // this lane's output column

    // ---------------- layer 1: h = relu(out @ W1 + b1) ----------------
    v8f acc;
    {
        const float bias = b1[cn];    // bias depends only on N
        #pragma unroll
        for (int i = 0; i < 8; ++i) acc[i] = bias;
    }
    #pragma unroll
    for (int k = 0; k < 32; ++k) {
        const int kb = 4 * k + 2 * half;
        v2f a = *(const v2f*)&s_in[l16 * LDS_STRIDE + kb];      // ds_load_b64
        v2f b;
        b.x = W1[(size_t)kb * D + cn];                          // L2-hot weights
        b.y = W1[(size_t)(kb + 1) * D + cn];
        acc = __builtin_amdgcn_wmma_f32_16x16x4_f32(
            /*neg_a=*/false, a, /*neg_b=*/false, b,
            /*c_mod=*/(short)0, acc, /*reuse_a=*/false, /*reuse_b=*/false);
    }
    #pragma unroll
    for (int r = 0; r < 8; ++r) {
        const int m = r + 8 * half;
        const float hv = acc[r] > 0.0f ? acc[r] : 0.0f;
        s_h[m * LDS_STRIDE + cn] = hv;
    }
    __syncthreads();

    // ---------------- layer 2: y = h @ W2 + b2 ----------------
    {
        const float bias = b2[cn];
        #pragma unroll
        for (int i = 0; i < 8; ++i) acc[i] = bias;
    }
    #pragma unroll
    for (int k = 0; k < 32; ++k) {
        const int kb = 4 * k + 2 * half;
        v2f a = *(const v2f*)&s_h[l16 * LDS_STRIDE + kb];
        v2f b;
        b.x = W2[(size_t)kb * D + cn];
        b.y = W2[(size_t)(kb + 1) * D + cn];
        acc = __builtin_amdgcn_wmma_f32_16x16x4_f32(
            false, a, false, b, (short)0, acc, false, false);
    }
    #pragma unroll
    for (int r = 0; r < 8; ++r) {
        const int m = r + 8 * half;
        out[(rowbase + m) * D + cn] = acc[r];
    }
}

// ---------------------------------------------------------------------------
extern "C" void kernel_launch(void* const* d_in, const int* in_sizes, int n_in,
                              void* d_out, int out_size, void* d_ws, size_t ws_size,
                              hipStream_t stream) {
    const float*     x   = (const float*)d_in[0];
    const long long* ei  = (const long long*)d_in[1];  // int64 edge_index [2, E]
    const float*     eps = (const float*)d_in[2];
    const float*     W1  = (const float*)d_in[3];
    const float*     b1  = (const float*)d_in[4];
    const float*     W2  = (const float*)d_in[5];
    const float*     b2  = (const float*)d_in[6];
    float* out = (float*)d_out;

    (void)in_sizes; (void)n_in; (void)out_size; (void)d_ws; (void)ws_size;

    // 1) out = (1+eps)*x          (12.8M floats -> 3.2M float4 -> 12500 blocks)
    gin_init<<<(N_NODES * D / 4) / 256, 256, 0, stream>>>(x, eps, out);

    // 2) scatter-add: 1.6M edges, 8 edges (waves) per 256-thread block
    gin_scatter<<<N_EDGES / 8, 256, 0, stream>>>(x, ei, out);

    // 3) fused 2-layer MLP via f32 WMMA, 16 rows per block (100000/16 = 6250)
    gin_mlp<<<N_NODES / 16, 256, 0, stream>>>(out, W1, b1, W2, b2);
}